// ChannelAttention_14448269984372
// MI455X (gfx1250) — compile-verified
//
#include <hip/hip_runtime.h>
#include <hip/hip_bf16.h>
#include <math.h>

typedef __attribute__((ext_vector_type(2))) float v2f;
typedef __attribute__((ext_vector_type(4))) float v4f;
typedef __attribute__((ext_vector_type(8))) float v8f;

__device__ __forceinline__ v8f wmma4(v2f a, v2f b, v8f c) {
  // V_WMMA_F32_16X16X4_F32 : D(16x16,f32) = A(16x4,f32) x B(4x16,f32) + C
  return __builtin_amdgcn_wmma_f32_16x16x4_f32(
      /*neg_a=*/false, a, /*neg_b=*/false, b,
      /*c_mod=*/(short)0, c, /*reuse_a=*/false, /*reuse_b=*/false);
}

// ---------------------------------------------------------------------------
// Pass 1: y[row] = mean over c=512 of x[row*512 .. row*512+511]
// One wave32 per row; 128-bit loads (lane L reads vec4 L, L+32, L+64, L+96).
// ---------------------------------------------------------------------------
__global__ void __launch_bounds__(256) k_reduce(const float* __restrict__ x,
                                                float* __restrict__ y,
                                                int nrows) {
  int wave = blockIdx.x * (blockDim.x >> 5) + (threadIdx.x >> 5);
  int lane = threadIdx.x & 31;
  if (wave >= nrows) return;
  const v4f* row = (const v4f*)(x + (size_t)wave * 512);
  float s = 0.0f;
#pragma unroll
  for (int i = 0; i < 4; ++i) {
    v4f v = row[lane + i * 32];
    s += (v.x + v.y) + (v.z + v.w);
  }
#pragma unroll
  for (int off = 16; off > 0; off >>= 1)
    s += __shfl_xor(s, off, 32);
  if (lane == 0) y[wave] = s * (1.0f / 512.0f);
}

// ---------------------------------------------------------------------------
// Pass 2: tiny MLP + softmax attention, all GEMMs on v_wmma_f32_16x16x4_f32.
// One wave per 16-row tile of the 2048-row batch.
//
// A-frag (16x4 f32, ISA 7.12.2): lanes 0-15 hold row m=lane, K={4kc,4kc+1};
//                                lanes 16-31 hold row m=lane-16, K={4kc+2,4kc+3}.
// B-frag (4x16): lanes 0-15 hold col n=lane, K={kb,kb+1} with kb=4kc (+2 hi half).
// D (16x16 f32, 8 VGPRs): reg r -> row r (lanes 0-15) / r+8 (lanes 16-31), col=lane&15.
// ---------------------------------------------------------------------------
__global__ void __launch_bounds__(32) k_attn(const float* __restrict__ y,   // (2048,64)
                                             const float* __restrict__ w1,  // (32,64)
                                             const float* __restrict__ b1,  // (32)
                                             const float* __restrict__ w2,  // (64,32)
                                             const float* __restrict__ b2,  // (64)
                                             const float* __restrict__ mb,  // (32,128)
                                             float* __restrict__ scale) {   // (2048,64)
  __shared__ float lrbuf[16 * 32];   // low_rank, later reused for y1
  __shared__ float fbuf[16 * 128];   // f1 / attn

  const int lane  = threadIdx.x;          // 0..31
  const int m     = lane & 15;            // row (A) or col (B/D) within tile
  const int khalf = (lane >> 4) * 2;      // 0 for lanes 0-15, 2 for lanes 16-31
  const int rhalf = (lane >> 4) << 3;     // D-row offset: 0 or 8
  const int r0    = blockIdx.x * 16;      // first batch row of this tile

  // ---- Stage 1: low_rank(16x32) = y_tile(16x64) @ w1^T(64x32) + b1
  for (int nt = 0; nt < 2; ++nt) {
    v8f acc = {};
    for (int kc = 0; kc < 16; ++kc) {
      int kb = kc * 4 + khalf;
      v2f a; a.x = y[(r0 + m) * 64 + kb];     a.y = y[(r0 + m) * 64 + kb + 1];
      int nn = nt * 16 + m;
      v2f b; b.x = w1[nn * 64 + kb];          b.y = w1[nn * 64 + kb + 1]; // w1^T[k][n]=w1[n][k]
      acc = wmma4(a, b, acc);
    }
    int col = nt * 16 + m;
    float bias = b1[col];
#pragma unroll
    for (int r = 0; r < 8; ++r)
      lrbuf[(r + rhalf) * 32 + col] = acc[r] + bias;
  }
  __syncthreads();

  // ---- Stage 2: f1(16x128) = low_rank(16x32) @ mb(32x128), scaled by 32^-0.5
  const float sm = 0.17677669529663687f;  // 1/sqrt(32)
  for (int nt = 0; nt < 8; ++nt) {
    v8f acc = {};
    for (int kc = 0; kc < 8; ++kc) {
      int kb = kc * 4 + khalf;
      v2f a; a.x = lrbuf[m * 32 + kb];        a.y = lrbuf[m * 32 + kb + 1];
      int nn = nt * 16 + m;
      v2f b; b.x = mb[kb * 128 + nn];         b.y = mb[(kb + 1) * 128 + nn];
      acc = wmma4(a, b, acc);
    }
    int col = nt * 16 + m;
#pragma unroll
    for (int r = 0; r < 8; ++r)
      fbuf[(r + rhalf) * 128 + col] = acc[r] * sm;
  }
  __syncthreads();

  // ---- Stage 3: row-wise softmax over 128 (lanes 0-15, one row each)
  if (lane < 16) {
    float mx = -1e30f;
    for (int j = 0; j < 128; ++j) mx = fmaxf(mx, fbuf[lane * 128 + j]);
    float sum = 0.0f;
    for (int j = 0; j < 128; ++j) {
      float e = expf(fbuf[lane * 128 + j] - mx);
      fbuf[lane * 128 + j] = e;
      sum += e;
    }
    float inv = 1.0f / sum;
    for (int j = 0; j < 128; ++j) fbuf[lane * 128 + j] *= inv;
  }
  __syncthreads();

  // ---- Stage 4: y1(16x32) = attn(16x128) @ mb^T(128x32)   (reuse lrbuf)
  for (int nt = 0; nt < 2; ++nt) {
    v8f acc = {};
    for (int kc = 0; kc < 32; ++kc) {
      int kb = kc * 4 + khalf;
      v2f a; a.x = fbuf[m * 128 + kb];        a.y = fbuf[m * 128 + kb + 1];
      int nn = nt * 16 + m;
      v2f b; b.x = mb[nn * 128 + kb];         b.y = mb[nn * 128 + kb + 1]; // mb^T[k][n]=mb[n][k]
      acc = wmma4(a, b, acc);
    }
    int col = nt * 16 + m;
#pragma unroll
    for (int r = 0; r < 8; ++r)
      lrbuf[(r + rhalf) * 32 + col] = acc[r];
  }
  __syncthreads();

  // ---- Stage 5: scale(16x64) = sigmoid(y1(16x32) @ w2^T(32x64) + b2)
  for (int nt = 0; nt < 4; ++nt) {
    v8f acc = {};
    for (int kc = 0; kc < 8; ++kc) {
      int kb = kc * 4 + khalf;
      v2f a; a.x = lrbuf[m * 32 + kb];        a.y = lrbuf[m * 32 + kb + 1];
      int nn = nt * 16 + m;
      v2f b; b.x = w2[nn * 32 + kb];          b.y = w2[nn * 32 + kb + 1]; // w2^T[k][n]=w2[n][k]
      acc = wmma4(a, b, acc);
    }
    int col = nt * 16 + m;
    float bias = b2[col];
#pragma unroll
    for (int r = 0; r < 8; ++r) {
      float v = acc[r] + bias;
      scale[(size_t)(r0 + r + rhalf) * 64 + col] = 1.0f / (1.0f + expf(-v));
    }
  }
}

// ---------------------------------------------------------------------------
// Pass 3: out = x * scale (scale broadcast over c=512). 128-bit streaming,
// nontemporal store (output never re-read by us).
// ---------------------------------------------------------------------------
__global__ void __launch_bounds__(256) k_scale(const float* __restrict__ x,
                                               const float* __restrict__ scale,
                                               float* __restrict__ out,
                                               size_t n4) {
  size_t i = (size_t)blockIdx.x * blockDim.x + threadIdx.x;
  if (i >= n4) return;
  v4f v = ((const v4f*)x)[i];
  float s = scale[i >> 7];                 // 128 vec4 per 512-float row
  v4f o = v * s;
  __builtin_nontemporal_store(o, (v4f*)out + i);
}

extern "C" void kernel_launch(void* const* d_in, const int* in_sizes, int n_in,
                              void* d_out, int out_size, void* d_ws, size_t ws_size,
                              hipStream_t stream) {
  (void)n_in; (void)out_size; (void)ws_size;
  const float* x  = (const float*)d_in[0];  // (2048, 64, 512) flattened
  const float* w1 = (const float*)d_in[1];  // (32, 64)
  const float* b1 = (const float*)d_in[2];  // (32)
  const float* w2 = (const float*)d_in[3];  // (64, 32)
  const float* b2 = (const float*)d_in[4];  // (64)
  const float* mb = (const float*)d_in[5];  // (32, 128)
  float* out = (float*)d_out;

  const int n = 64, c = 512;
  const int B = in_sizes[0] / (n * c);      // 2048 for reference shapes

  float* y     = (float*)d_ws;              // B*n floats
  float* scale = (float*)d_ws + (size_t)B * n;

  // Pass 1: per-(B,n) mean over c. One wave per row, 8 waves per block.
  int nrows = B * n;                        // 131072
  int blocks1 = (nrows + 7) / 8;
  k_reduce<<<blocks1, 256, 0, stream>>>(x, y, nrows);

  // Pass 2: WMMA MLP + softmax attention. One wave per 16-row tile.
  k_attn<<<B / 16, 32, 0, stream>>>(y, w1, b1, w2, b2, mb, scale);

  // Pass 3: broadcast multiply.
  size_t n4 = (size_t)B * n * c / 4;        // 16,777,216 vec4
  k_scale<<<(unsigned)((n4 + 255) / 256), 256, 0, stream>>>(x, scale, out, n4);
}